// Sampler_63634235458088
// MI455X (gfx1250) — compile-verified
//
#include <hip/hip_runtime.h>
#include <math.h>

// ---------------- problem constants (match reference) ----------------
#define NSEQ      256
#define VOCAB     128000
#define V4        (VOCAB / 4)
#define PLEN      512
#define OLEN      128
#define MASKW     (VOCAB / 32)      // 4000 dwords = 16 KB per bitmask
#define NBINS     2048
#define BIN_RANGE 16.0f             // histogram covers [rowmax-16, rowmax]
#define INV_BINW  ((float)NBINS / BIN_RANGE)
#define TILE      1024              // floats per async LDS tile (4 KB)
#define NTILES    (VOCAB / TILE)    // 125 exact tiles
#define NTHREADS  512               // 16 wave32
#define NWAVES    (NTHREADS / 32)

typedef float f32x4 __attribute__((ext_vector_type(4)));
typedef int   i32x4 __attribute__((ext_vector_type(4)));

// ---------------- CDNA5 async global->LDS path ----------------
#if __has_builtin(__builtin_amdgcn_global_load_async_to_lds_b128)
#define HAVE_ASYNC_LDS 1
#endif

__device__ __forceinline__ void async_copy16(const float* g, float* l) {
#ifdef HAVE_ASYNC_LDS
  __builtin_amdgcn_global_load_async_to_lds_b128(
      (__attribute__((address_space(1))) i32x4*)(g),
      (__attribute__((address_space(3))) i32x4*)(l),
      /*offset=*/0, /*cpol=*/0);
#else
  *(f32x4*)l = *(const f32x4*)g;
#endif
}

__device__ __forceinline__ void wait_async_le1() {
#ifdef HAVE_ASYNC_LDS
#if __has_builtin(__builtin_amdgcn_s_wait_asynccnt)
  __builtin_amdgcn_s_wait_asynccnt(1);
#else
  asm volatile("s_wait_asynccnt 1" ::: "memory");
#endif
#endif
}

__device__ __forceinline__ void wait_async_le0() {
#ifdef HAVE_ASYNC_LDS
#if __has_builtin(__builtin_amdgcn_s_wait_asynccnt)
  __builtin_amdgcn_s_wait_asynccnt(0);
#else
  asm volatile("s_wait_asynccnt 0" ::: "memory");
#endif
#endif
}

// ---------------- fused penalty application ----------------
__device__ __forceinline__ float penalize(float x, int v,
                                          const unsigned* __restrict__ pm,
                                          const unsigned* __restrict__ om,
                                          const int* __restrict__ otok,
                                          const unsigned char* __restrict__ ocnt,
                                          float rp, float fp, float pp) {
  const bool inout = (om[v >> 5] >> (v & 31)) & 1u;
  const bool seen  = inout || (((pm[v >> 5] >> (v & 31)) & 1u) != 0u);
  if (seen) x = (x > 0.0f) ? (x / rp) : (x * rp);
  if (inout) {                      // rare: <=128 of 128000 tokens per row
    int c = 1;
#pragma unroll 1
    for (int j = 0; j < OLEN; ++j) {
      if (otok[j] == v) { c = (int)ocnt[j]; break; }
    }
    x -= fp * (float)c + pp;        // frequency * count + presence
  }
  return x;
}

// ---------------- fused sampler kernel: one workgroup per row ----------------
__global__ void __launch_bounds__(NTHREADS)
sampler_fused_kernel(const float* __restrict__ logits,
                     const int*   __restrict__ prompt_tokens,
                     const int*   __restrict__ output_tokens,
                     const float* __restrict__ presence_p,
                     const float* __restrict__ frequency_p,
                     const float* __restrict__ repetition_p,
                     const float* __restrict__ top_p,
                     const int*   __restrict__ top_k,
                     const float* __restrict__ min_p,
                     float*       __restrict__ out) {
  __shared__ unsigned      s_pmask[MASKW];     // prompt token bitmask (16 KB)
  __shared__ unsigned      s_omask[MASKW];     // output token bitmask (16 KB)
  __shared__ int           s_otok[OLEN];
  __shared__ unsigned char s_ocnt[OLEN];
  __shared__ unsigned      s_hcnt[NBINS];      // 8 KB
  __shared__ float         s_hexp[NBINS];      // 8 KB
  __shared__ float         s_red[NWAVES];
  __shared__ float         s_tile[2][TILE];    // 8 KB double buffer
  __shared__ float         s_max, s_thresh;

  const int row = blockIdx.x;
  const int tid = threadIdx.x;

  const float rp = repetition_p[row];
  const float fp = frequency_p[row];
  const float pp = presence_p[row];
  const float tp = top_p[row];
  const int   tk = top_k[row];
  const float mp = min_p[row];

  const float* rowlog = logits + (size_t)row * VOCAB;
  float*       rowout = out    + (size_t)row * VOCAB;

  // -------- phase 0: build token masks / counts in LDS --------
  for (int i = tid; i < MASKW; i += NTHREADS) { s_pmask[i] = 0u; s_omask[i] = 0u; }
  for (int i = tid; i < NBINS; i += NTHREADS) { s_hcnt[i] = 0u; s_hexp[i] = 0.0f; }
  __syncthreads();

  for (int i = tid; i < PLEN; i += NTHREADS) {
    const int t = prompt_tokens[row * PLEN + i];
    atomicOr(&s_pmask[t >> 5], 1u << (t & 31));     // ds_or_b32
  }
  if (tid < OLEN) {
    const int t = output_tokens[row * OLEN + tid];
    s_otok[tid] = t;
    atomicOr(&s_omask[t >> 5], 1u << (t & 31));
  }
  __syncthreads();
  if (tid < OLEN) {
    const int t = s_otok[tid];
    int c = 0;
#pragma unroll 1
    for (int j = 0; j < OLEN; ++j) c += (s_otok[j] == t);
    s_ocnt[tid] = (unsigned char)c;
  }
  __syncthreads();

  // -------- phase 1: penalized row max (b128 vector loads) --------
  float lmax = -__builtin_inff();
  for (int i = tid; i < V4; i += NTHREADS) {
    const f32x4 x = ((const f32x4*)rowlog)[i];
    const int v = i << 2;
    lmax = fmaxf(lmax, penalize(x.x, v + 0, s_pmask, s_omask, s_otok, s_ocnt, rp, fp, pp));
    lmax = fmaxf(lmax, penalize(x.y, v + 1, s_pmask, s_omask, s_otok, s_ocnt, rp, fp, pp));
    lmax = fmaxf(lmax, penalize(x.z, v + 2, s_pmask, s_omask, s_otok, s_ocnt, rp, fp, pp));
    lmax = fmaxf(lmax, penalize(x.w, v + 3, s_pmask, s_omask, s_otok, s_ocnt, rp, fp, pp));
  }
  for (int off = 16; off > 0; off >>= 1) lmax = fmaxf(lmax, __shfl_down(lmax, off, 32));
  if ((tid & 31) == 0) s_red[tid >> 5] = lmax;
  __syncthreads();
  if (tid == 0) {
    float m = -__builtin_inff();
    for (int w = 0; w < NWAVES; ++w) m = fmaxf(m, s_red[w]);
    s_max = m;
  }
  __syncthreads();
  const float rmax = s_max;

  // -------- phase 2: LDS histogram via double-buffered async tiles --------
  if (tid < TILE / 4)                       // waves 0..7: 1 async b128 per wave/tile
    async_copy16(rowlog + tid * 4, &s_tile[0][tid * 4]);

  for (int t = 0; t < NTILES; ++t) {
    const int buf = t & 1;
    if (t + 1 < NTILES) {
      if (tid < TILE / 4)
        async_copy16(rowlog + (t + 1) * TILE + tid * 4, &s_tile[buf ^ 1][tid * 4]);
      wait_async_le1();                     // tile t done (in-order), t+1 in flight
    } else {
      wait_async_le0();
    }
    __syncthreads();
    const int base = t * TILE;
    for (int j = tid; j < TILE; j += NTHREADS) {
      const float px = penalize(s_tile[buf][j], base + j,
                                s_pmask, s_omask, s_otok, s_ocnt, rp, fp, pp);
      const float e = __expf(px - rmax);
      int bin = (int)((rmax - px) * INV_BINW);
      bin = bin < 0 ? 0 : (bin > NBINS - 1 ? NBINS - 1 : bin);
      atomicAdd(&s_hcnt[bin], 1u);          // ds_add_u32
      atomicAdd(&s_hexp[bin], e);           // ds_add_f32
    }
    __syncthreads();
  }

  // -------- phase 2b: softmax total + bin scan -> combined threshold --------
  float tsum = 0.0f;
  for (int i = tid; i < NBINS; i += NTHREADS) tsum += s_hexp[i];
  for (int off = 16; off > 0; off >>= 1) tsum += __shfl_down(tsum, off, 32);
  if ((tid & 31) == 0) s_red[tid >> 5] = tsum;
  __syncthreads();
  if (tid == 0) {
    float T = 0.0f;
    for (int w = 0; w < NWAVES; ++w) T += s_red[w];
    const float pT = tp * T;                // exclusive prob-mass budget
    unsigned cum_c = 0;
    float cum_e = 0.0f;
    int bk = NBINS - 1, bp = 0;
    bool foundk = false;
#pragma unroll 1
    for (int b = 0; b < NBINS; ++b) {
      if (cum_e <= pT) bp = b;              // bin starts within top-p budget
      cum_c += s_hcnt[b];
      cum_e += s_hexp[b];
      if (!foundk && cum_c >= (unsigned)tk) { bk = b; foundk = true; }
    }
    const int bstar = bk < bp ? bk : bp;
    const float thr_kp   = rmax - (float)(bstar + 1) * (BIN_RANGE / (float)NBINS);
    const float thr_minp = rmax + __logf(mp);   // min_p==0 -> -inf -> no-op
    s_thresh = fmaxf(thr_kp, thr_minp);
  }
  __syncthreads();
  const float thr = s_thresh;

  // -------- phase 3: apply mask, non-temporal b128 stores --------
  for (int i = tid; i < V4; i += NTHREADS) {
    const f32x4 x = ((const f32x4*)rowlog)[i];
    const int v = i << 2;
    f32x4 y;
    float p0 = penalize(x.x, v + 0, s_pmask, s_omask, s_otok, s_ocnt, rp, fp, pp);
    float p1 = penalize(x.y, v + 1, s_pmask, s_omask, s_otok, s_ocnt, rp, fp, pp);
    float p2 = penalize(x.z, v + 2, s_pmask, s_omask, s_otok, s_ocnt, rp, fp, pp);
    float p3 = penalize(x.w, v + 3, s_pmask, s_omask, s_otok, s_ocnt, rp, fp, pp);
    y.x = (p0 >= thr) ? p0 : -__builtin_inff();
    y.y = (p1 >= thr) ? p1 : -__builtin_inff();
    y.z = (p2 >= thr) ? p2 : -__builtin_inff();
    y.w = (p3 >= thr) ? p3 : -__builtin_inff();
    __builtin_nontemporal_store(y, &((f32x4*)rowout)[i]);
  }
}

// ---------------- launch ----------------
extern "C" void kernel_launch(void* const* d_in, const int* in_sizes, int n_in,
                              void* d_out, int out_size, void* d_ws, size_t ws_size,
                              hipStream_t stream) {
  (void)in_sizes; (void)n_in; (void)out_size; (void)d_ws; (void)ws_size;
  const float* logits        = (const float*)d_in[0];
  const int*   prompt_tokens = (const int*)  d_in[1];
  const int*   output_tokens = (const int*)  d_in[2];
  const float* presence      = (const float*)d_in[3];
  const float* frequency     = (const float*)d_in[4];
  const float* repetition    = (const float*)d_in[5];
  const float* topp          = (const float*)d_in[6];
  const int*   topk          = (const int*)  d_in[7];
  const float* minp          = (const float*)d_in[8];
  float*       out           = (float*)d_out;

  sampler_fused_kernel<<<dim3(NSEQ), dim3(NTHREADS), 0, stream>>>(
      logits, prompt_tokens, output_tokens, presence, frequency, repetition,
      topp, topk, minp, out);
}